// AspectModel_31095563223834
// MI455X (gfx1250) — compile-verified
//
#include <hip/hip_runtime.h>
#include <hip/hip_bf16.h>

typedef float v2f __attribute__((ext_vector_type(2)));
typedef float v8f __attribute__((ext_vector_type(8)));

// Problem constants (fixed by the reference):
//   B=32, S=256, H=256, D=2H=512, 3D=1536, 4D=2048, F=1024, T=5
#define SLEN   256
#define DDIM   512
#define D3     1536
#define TAGS   5
#define FB     16          // fragments per block
#define NTHR   256         // 8 wave32 waves
#define PSPAN  1540        // span row pitch (floats): 1540 % 64 == 4 -> conflict-free A reads
#define PV     516         // v row pitch (floats): 516*4 bytes is 16B aligned for float4

__global__ __launch_bounds__(NTHR)
void aspect_kernel(const float* __restrict__ en,      // [B,S,D]
                   const int*   __restrict__ lengths, // [B]
                   const int*   __restrict__ fb,      // [F]
                   const int*   __restrict__ fs,      // [F]
                   const int*   __restrict__ fe,      // [F]
                   const float* __restrict__ attw,    // [3D, D] row-major
                   const float* __restrict__ attb,    // [3D]
                   const float* __restrict__ tagw,    // [T, 4D]
                   const float* __restrict__ tagb,    // [T]
                   float*       __restrict__ out,     // [F, T]
                   int F)
{
    __shared__ float s_span[FB * PSPAN];   // 98,560 B  span = [l | word_state | r]
    __shared__ float s_v   [FB * PV];      // 33,024 B  v = span @ att_w
    __shared__ float s_att [FB * SLEN];    // 16,384 B  scores -> att*pos_weight
    __shared__ float s_part[8 * TAGS];     // warp partials (also reused for c reduction)
    __shared__ float s_c   [FB];
    __shared__ float s_log [TAGS];
    __shared__ int   s_b[FB], s_s0[FB], s_e[FB], s_len[FB];

    const int tid   = threadIdx.x;
    const int lane  = tid & 31;
    const int wave  = tid >> 5;
    const int fbase = blockIdx.x * FB;

    // ---- fragment metadata -------------------------------------------------
    if (tid < FB) {
        int f = fbase + tid;
        int b = 0, s0 = 0, e = 1;
        if (f < F) { b = fb[f]; s0 = fs[f]; e = fe[f]; }
        s_b[tid] = b; s_s0[tid] = s0; s_e[tid] = e; s_len[tid] = lengths[b];
    }
    __syncthreads();
    const float seq_len = (float)lengths[0];

    // ---- Phase 1: build span vectors in LDS --------------------------------
    for (int fi = 0; fi < FB; ++fi) {
        const int b = s_b[fi], s0 = s_s0[fi], e = s_e[fi];
        const float* base = en + (size_t)b * SLEN * DDIM;
        float* sp = s_span + fi * PSPAN;
        #pragma unroll
        for (int j = 0; j < 2; ++j) {
            const int d = tid + j * NTHR;
            float l  = base[(size_t)s0 * DDIM + d];
            float r  = base[(size_t)(e - 1) * DDIM + d];
            float ws = 0.f;
            for (int s = s0; s < e; ++s) ws += base[(size_t)s * DDIM + d];
            sp[d] = l; sp[DDIM + d] = ws; sp[2 * DDIM + d] = r;
        }
    }
    __syncthreads();

    // ---- Phase 2: c[fi] = span . att_b -------------------------------------
    for (int fi = 0; fi < FB; ++fi) {
        const float* sp = s_span + fi * PSPAN;
        float p = 0.f;
        #pragma unroll
        for (int j = 0; j < 6; ++j) { int k = tid + j * NTHR; p += sp[k] * attb[k]; }
        #pragma unroll
        for (int off = 16; off; off >>= 1) p += __shfl_xor(p, off, 32);
        if (lane == 0) s_part[wave] = p;
        __syncthreads();
        if (tid == 0) {
            float c = 0.f;
            #pragma unroll
            for (int w = 0; w < 8; ++w) c += s_part[w];
            s_c[fi] = c;
        }
        __syncthreads();
    }

    // ---- Phase 3: v = span @ att_w via V_WMMA_F32_16X16X4_F32 --------------
    // A = span  [M=16 frag, K=1536] from LDS
    // B = att_w [K=1536, N=512] from global (L2-resident, shared by all blocks)
    // Each wave owns 4 contiguous 16-wide N tiles.
    {
        const int half = lane >> 4;        // 0: K=0,1   1: K=2,3
        const int m    = lane & 15;        // A row (fragment) / B column-in-tile
        const float* sp = s_span + m * PSPAN;
        const int n0 = wave * 64;          // 4 tiles x 16 columns

        v8f acc0 = {}, acc1 = {}, acc2 = {}, acc3 = {};
        #pragma unroll 2
        for (int k = 0; k < D3; k += 4) {
            v2f a;
            a.x = sp[k + 2 * half];
            a.y = sp[k + 2 * half + 1];
            const float* bw0 = attw + (size_t)(k + 2 * half) * DDIM + n0 + m;
            const float* bw1 = bw0 + DDIM;
            v2f b0; b0.x = bw0[ 0]; b0.y = bw1[ 0];
            v2f b1; b1.x = bw0[16]; b1.y = bw1[16];
            v2f b2; b2.x = bw0[32]; b2.y = bw1[32];
            v2f b3; b3.x = bw0[48]; b3.y = bw1[48];
            acc0 = __builtin_amdgcn_wmma_f32_16x16x4_f32(false, a, false, b0, (short)0, acc0, false, false);
            acc1 = __builtin_amdgcn_wmma_f32_16x16x4_f32(false, a, false, b1, (short)0, acc1, false, false);
            acc2 = __builtin_amdgcn_wmma_f32_16x16x4_f32(false, a, false, b2, (short)0, acc2, false, false);
            acc3 = __builtin_amdgcn_wmma_f32_16x16x4_f32(false, a, false, b3, (short)0, acc3, false, false);
        }
        // C layout: VGPR r, lanes 0-15 -> M=r, lanes 16-31 -> M=8+r, N = lane&15
        #pragma unroll
        for (int r = 0; r < 8; ++r) {
            const int mm = r + 8 * half;
            float* vd = s_v + mm * PV + n0 + m;
            vd[ 0] = acc0[r];
            vd[16] = acc1[r];
            vd[32] = acc2[r];
            vd[48] = acc3[r];
        }
    }
    __syncthreads();

    // ---- Phase 4: scores[fi][s] = tanh(pw * (mem[s].v) + c), masked --------
    for (int fi = 0; fi < FB; ++fi) {
        const int b = s_b[fi], s0 = s_s0[fi], e = s_e[fi], len = s_len[fi];
        const int s = tid;                               // one thread per position
        float score;
        const bool masked = (s >= len) || (s >= s0 && s < e);
        if (masked) {
            score = -1e4f;
        } else {
            const float4* mem = (const float4*)(en + ((size_t)b * SLEN + s) * DDIM);
            const float4* vv  = (const float4*)(s_v + fi * PV);
            float dot = 0.f;
            #pragma unroll 4
            for (int q = 0; q < DDIM / 4; ++q) {
                float4 mq = mem[q], vq = vv[q];
                dot += mq.x * vq.x + mq.y * vq.y + mq.z * vq.z + mq.w * vq.w;
            }
            const int dis = (s < s0) ? (s0 - s) : (s - e + 1);
            const float pw = 1.f - (float)dis / seq_len;
            score = tanhf(pw * dot + s_c[fi]);
        }
        s_att[fi * SLEN + s] = score;
    }
    __syncthreads();

    // ---- Phase 5: softmax per fragment; fold pos_weight into weights -------
    #pragma unroll
    for (int t = 0; t < 2; ++t) {
        const int fi = wave * 2 + t;
        float* sc = s_att + fi * SLEN;
        const int s0 = s_s0[fi], e = s_e[fi];

        float mx = -3.4e38f;
        for (int i = lane; i < SLEN; i += 32) mx = fmaxf(mx, sc[i]);
        #pragma unroll
        for (int off = 16; off; off >>= 1) mx = fmaxf(mx, __shfl_xor(mx, off, 32));

        float sum = 0.f;
        for (int i = lane; i < SLEN; i += 32) { float ex = __expf(sc[i] - mx); sc[i] = ex; sum += ex; }
        #pragma unroll
        for (int off = 16; off; off >>= 1) sum += __shfl_xor(sum, off, 32);
        const float inv = 1.f / sum;

        for (int i = lane; i < SLEN; i += 32) {
            float pw;
            if (i >= s0 && i < e) pw = 0.f;                       // pos_dis = seq_len -> pw = 0
            else { int dis = (i < s0) ? (s0 - i) : (i - e + 1); pw = 1.f - (float)dis / seq_len; }
            sc[i] = sc[i] * inv * pw;                             // att * pos_weight
        }
    }
    __syncthreads();

    // ---- Phase 6: mix + logits + log_softmax (fused) -----------------------
    for (int fi = 0; fi < FB; ++fi) {
        const int b = s_b[fi];
        const float* w    = s_att + fi * SLEN;
        const float* base = en + (size_t)b * SLEN * DDIM;

        float m0 = 0.f, m1 = 0.f;                  // mix[d=tid], mix[d=tid+256]
        for (int s = 0; s < SLEN; ++s) {
            const float ws = w[s];
            m0 += ws * base[(size_t)s * DDIM + tid];
            m1 += ws * base[(size_t)s * DDIM + tid + NTHR];
        }

        const float* sp = s_span + fi * PSPAN;
        float p[TAGS];
        #pragma unroll
        for (int t = 0; t < TAGS; ++t) {
            const float* tw = tagw + t * (4 * DDIM);
            float acc = m0 * tw[D3 + tid] + m1 * tw[D3 + tid + NTHR];
            #pragma unroll
            for (int j = 0; j < 6; ++j) { int k = tid + j * NTHR; acc += sp[k] * tw[k]; }
            p[t] = acc;
        }
        #pragma unroll
        for (int t = 0; t < TAGS; ++t)
            #pragma unroll
            for (int off = 16; off; off >>= 1) p[t] += __shfl_xor(p[t], off, 32);
        if (lane == 0)
            #pragma unroll
            for (int t = 0; t < TAGS; ++t) s_part[wave * TAGS + t] = p[t];
        __syncthreads();

        if (tid < TAGS) {
            float lg = tagb[tid];
            #pragma unroll
            for (int w8 = 0; w8 < 8; ++w8) lg += s_part[w8 * TAGS + tid];
            s_log[tid] = lg;
        }
        __syncthreads();

        const int f = fbase + fi;
        if (tid == 0 && f < F) {
            float mx = s_log[0];
            #pragma unroll
            for (int t = 1; t < TAGS; ++t) mx = fmaxf(mx, s_log[t]);
            float sum = 0.f;
            #pragma unroll
            for (int t = 0; t < TAGS; ++t) sum += __expf(s_log[t] - mx);
            const float lse = mx + __logf(sum);
            #pragma unroll
            for (int t = 0; t < TAGS; ++t) out[(size_t)f * TAGS + t] = s_log[t] - lse;
        }
        __syncthreads();
    }
}

extern "C" void kernel_launch(void* const* d_in, const int* in_sizes, int n_in,
                              void* d_out, int out_size, void* d_ws, size_t ws_size,
                              hipStream_t stream) {
    const float* en      = (const float*)d_in[0];
    const int*   lengths = (const int*)  d_in[1];
    const int*   fbp     = (const int*)  d_in[2];
    const int*   fsp     = (const int*)  d_in[3];
    const int*   fep     = (const int*)  d_in[4];
    const float* attw    = (const float*)d_in[5];
    const float* attb    = (const float*)d_in[6];
    const float* tagw    = (const float*)d_in[7];
    const float* tagb    = (const float*)d_in[8];
    float*       out     = (float*)d_out;

    const int F = in_sizes[2];
    const int blocks = (F + FB - 1) / FB;
    aspect_kernel<<<blocks, NTHR, 0, stream>>>(en, lengths, fbp, fsp, fep,
                                               attw, attb, tagw, tagb, out, F);
}